// scattering_80410377716428
// MI455X (gfx1250) — compile-verified
//
#include <hip/hip_runtime.h>
#include <math.h>

// Problem constants (match reference)
#define NN   2048
#define DF   256
#define JF   8          // filter bank size
#define KC   32         // Chebyshev order
#define ND   (NN * DF)  // elements per N x DF matrix

// GEMM tiling
#define KCH  32                 // K-chunk staged in LDS
#define NCH  (NN / KCH)         // 64 chunks
#define ASTR 36                 // padded LDS row stride for A tile (floats, x4 -> 16B aligned)
#define BSTR 72                 // padded LDS row stride for B tile (floats)

typedef __attribute__((ext_vector_type(2))) float v2f;
typedef __attribute__((ext_vector_type(8))) float v8f;

#if defined(__has_builtin)
#if __has_builtin(__builtin_amdgcn_global_load_async_to_lds_b128)
#define ASYNC_LDS 1
#endif
#endif

#if defined(ASYNC_LDS)
// Exact parameter types per hipcc diagnostics:
//   arg0: global-AS (1) pointer to 16B int vector
//   arg1: LDS-AS   (3) pointer to 16B int vector
typedef int async_v4i __attribute__((vector_size(16)));
typedef __attribute__((address_space(1))) async_v4i* gbl_v4i_ptr;
typedef __attribute__((address_space(1))) void*      gbl_void_ptr;
typedef __attribute__((address_space(3))) async_v4i* lds_v4i_ptr;
typedef __attribute__((address_space(3))) void*      lds_void_ptr;
#endif

__device__ __forceinline__ void wait_async0() {
#if defined(ASYNC_LDS)
#if __has_builtin(__builtin_amdgcn_s_wait_asynccnt)
    __builtin_amdgcn_s_wait_asynccnt(0);
#else
    asm volatile("s_wait_asynccnt 0x0" ::: "memory");
#endif
#endif
}

__device__ __forceinline__ void stage16(const float* gp, float* lp) {
#if defined(ASYNC_LDS)
    __builtin_amdgcn_global_load_async_to_lds_b128(
        (gbl_v4i_ptr)(gbl_void_ptr)(void*)gp,
        (lds_v4i_ptr)(lds_void_ptr)lp, 0, 0);
#else
    float4 v = *(const float4*)gp;
    *(float4*)lp = v;
#endif
}

// ---------------------------------------------------------------------------
// Filter math (device) : A = R*ln(2)/(J-R+1) = 3*ln2/6 = ln2/2
// ---------------------------------------------------------------------------
__device__ __forceinline__ float ghat_dev(float x) {
    const float Aa = 0.34657359028f;
    float ind = (x <= 0.0f && x > -Aa) ? 1.0f : 0.0f;
    // d0 + d1*cos(2*pi*(x/A + 0.5)) with d0 = d1 = 0.5
    float val = 0.5f + 0.5f * cosf(6.28318530718f * (x / Aa + 0.5f));
    return val * ind;
}

__device__ __forceinline__ float wav_dev(float lam, int j) {  // j in [2,8]
    const float Aa = 0.34657359028f;
    const float Rr = 3.0f;
    float sh   = Aa * (float)(j - 1) / Rr;
    float lmin = expf(sh - Aa);
    float l    = fmaxf(lam, lmin);
    return ghat_dev(logf(l) - sh);
}

__device__ __forceinline__ float filt_dev(int jf, float lam) {
    if (jf == 0) {
        // scaling: sqrt(max(R*d0^2 + R/2*(d0^2+d1^2) - sum_j wav^2, 0)) = sqrt(max(1.5 - g3, 0))
        float g3 = 0.0f;
        #pragma unroll
        for (int j = 2; j <= 8; ++j) { float w = wav_dev(lam, j); g3 += w * w; }
        return sqrtf(fmaxf(1.5f - g3, 0.0f));
    }
    return wav_dev(lam, jf + 1);
}

// Chebyshev coefficients c[jf][k] for h_jf(1 + cos(theta)), Gauss-Chebyshev quadrature.
__global__ void cheb_coeffs_k(float* __restrict__ coef) {
    int jf = blockIdx.x;
    int k  = threadIdx.x;
    if (k >= KC) return;
    const int M = 512;
    float s = 0.0f;
    for (int m = 0; m < M; ++m) {
        float th  = 3.14159265359f * ((float)m + 0.5f) / (float)M;
        float lam = 1.0f + cosf(th);
        s += filt_dev(jf, lam) * cosf((float)k * th);
    }
    coef[jf * KC + k] = s * ((k == 0) ? 1.0f : 2.0f) / (float)M;
}

// ---------------------------------------------------------------------------
// Degrees + inverse sqrt degrees (one block per row, LDS reduction)
// ---------------------------------------------------------------------------
__global__ __launch_bounds__(256) void row_deg_k(const float* __restrict__ W,
                                                 float* __restrict__ deg,
                                                 float* __restrict__ dh) {
    __shared__ float red[256];
    int r = blockIdx.x;
    const float* row = W + (size_t)r * NN;
    float s = 0.0f;
    for (int c = threadIdx.x; c < NN; c += 256) s += row[c];
    red[threadIdx.x] = s;
    __syncthreads();
    for (int st = 128; st > 0; st >>= 1) {
        if (threadIdx.x < st) red[threadIdx.x] += red[threadIdx.x + st];
        __syncthreads();
    }
    if (threadIdx.x == 0) {
        deg[r] = red[0];
        dh[r]  = rsqrtf(fmaxf(red[0], 1.0f));
    }
}

// Build SHIFTED normalized Laplacian  Ls = (D^-1/2 (D - W) D^-1/2) - I
// (shift by -I maps spectrum [0,2] -> [-1,1] for the Chebyshev recurrence)
__global__ __launch_bounds__(256) void build_lap_k(const float* __restrict__ W,
                                                   const float* __restrict__ deg,
                                                   const float* __restrict__ dh,
                                                   float* __restrict__ Ls) {
    size_t i = (size_t)blockIdx.x * blockDim.x + threadIdx.x;
    if (i >= (size_t)NN * NN) return;
    int r = (int)(i / NN);
    int c = (int)(i % NN);
    float v = -dh[r] * W[i] * dh[c];
    if (r == c) v += deg[r] * dh[r] * dh[r] - 1.0f;
    Ls[i] = v;
}

// ---------------------------------------------------------------------------
// f32 WMMA GEMM:  Cout(2048x256) = alpha * A(2048x2048) @ B(2048x256) + beta * Cin
// Block: 256 thr = 8 waves, tile 128 rows x 64 cols; wave = 16 x 64 (4 accs).
// Double-buffered LDS staging of A (128xKCH) and B (KCHx64), coalesced b128
// async-to-LDS loads (ASYNCcnt-tracked), padded strides for conflict-free
// ds fragment reads.  Grid (16, 4).
// ---------------------------------------------------------------------------
__global__ __launch_bounds__(256) void gemm_f32_wmma(const float* __restrict__ A,
                                                     const float* __restrict__ B,
                                                     const float* __restrict__ Cin,
                                                     float* __restrict__ Cout,
                                                     float alpha, float beta) {
    __shared__ float As[2][128 * ASTR];   // 2 x 18 KB
    __shared__ float Bs[2][KCH * BSTR];   // 2 x  9 KB

    const int tid  = threadIdx.x;
    const int lane = tid & 31;
    const int wave = tid >> 5;
    const int mBlk = blockIdx.x * 128;
    const int n0   = blockIdx.y * 64;
    const int half = lane >> 4;     // 0 or 1
    const int l16  = lane & 15;
    const int kb   = half * 2;      // K-sub-offset for this half-wave

    v8f acc[4] = {};

    // Cooperative staging of one K-chunk into LDS buffer `buf`.
    auto stage = [&](int chunk, int buf) {
        const int ko = chunk * KCH;
        // A tile: 128 rows x 32 K  = 1024 float4, 4 per thread (coalesced 128B rows)
        #pragma unroll
        for (int i = 0; i < 4; ++i) {
            int idx = tid + i * 256;
            int r   = idx >> 3;
            int c4  = (idx & 7) * 4;
            stage16(A + (size_t)(mBlk + r) * NN + ko + c4, &As[buf][r * ASTR + c4]);
        }
        // B tile: 32 rows x 64 cols = 512 float4, 2 per thread
        #pragma unroll
        for (int i = 0; i < 2; ++i) {
            int idx = tid + i * 256;
            int r   = idx >> 4;
            int c4  = (idx & 15) * 4;
            stage16(B + (size_t)(ko + r) * DF + n0 + c4, &Bs[buf][r * BSTR + c4]);
        }
    };

    // WMMA over one staged K-chunk.
    auto compute = [&](int buf) {
        const float* Asw = &As[buf][(wave * 16 + l16) * ASTR];
        #pragma unroll
        for (int kk = 0; kk < KCH; kk += 4) {
            v2f a = *(const v2f*)(Asw + kk + kb);                 // 8B-aligned ds_load
            const float* Bp = &Bs[buf][(kk + kb) * BSTR + l16];
            v2f b0 = { Bp[0],  Bp[BSTR] };
            v2f b1 = { Bp[16], Bp[16 + BSTR] };
            v2f b2 = { Bp[32], Bp[32 + BSTR] };
            v2f b3 = { Bp[48], Bp[48 + BSTR] };
            acc[0] = __builtin_amdgcn_wmma_f32_16x16x4_f32(false, a, false, b0, (short)0, acc[0], false, false);
            acc[1] = __builtin_amdgcn_wmma_f32_16x16x4_f32(false, a, false, b1, (short)0, acc[1], false, false);
            acc[2] = __builtin_amdgcn_wmma_f32_16x16x4_f32(false, a, false, b2, (short)0, acc[2], false, false);
            acc[3] = __builtin_amdgcn_wmma_f32_16x16x4_f32(false, a, false, b3, (short)0, acc[3], false, false);
        }
    };

    stage(0, 0);
    wait_async0();
    __syncthreads();
    for (int c = 0; c < NCH; ++c) {
        int buf = c & 1;
        if (c + 1 < NCH) {
            stage(c + 1, buf ^ 1);
            if (c + 2 < NCH)   // pull the chunk after next toward L2
                __builtin_prefetch(A + (size_t)(mBlk + (tid >> 3)) * NN + (size_t)(c + 2) * KCH, 0, 3);
        }
        compute(buf);
        wait_async0();     // drain this wave's async LDS writes (other buffer)
        __syncthreads();   // all waves staged + done reading before overwrite
    }

    // C/D layout: VGPR v, lane -> row = m0 + 8*half + v, col = n0 + c*16 + l16
    const int m0 = mBlk + wave * 16;
    #pragma unroll
    for (int c = 0; c < 4; ++c) {
        #pragma unroll
        for (int v = 0; v < 8; ++v) {
            int    row = m0 + half * 8 + v;
            size_t idx = (size_t)row * DF + n0 + c * 16 + l16;
            float  val = alpha * acc[c][v];
            if (beta != 0.0f) val += beta * Cin[idx];
            Cout[idx] = val;
        }
    }
}

// ---------------------------------------------------------------------------
// Chebyshev accumulation helpers
// ---------------------------------------------------------------------------
// Y_j = c[j][0]*T0 + c[j][1]*T1  for j = 0..7
__global__ __launch_bounds__(256) void yinit_k(const float* __restrict__ T0,
                                               const float* __restrict__ T1,
                                               float* __restrict__ Y,
                                               const float* __restrict__ coef) {
    size_t i = (size_t)blockIdx.x * blockDim.x + threadIdx.x;
    if (i >= (size_t)ND) return;
    float t0 = T0[i], t1 = T1[i];
    #pragma unroll
    for (int j = 0; j < JF; ++j)
        Y[(size_t)j * ND + i] = coef[j * KC + 0] * t0 + coef[j * KC + 1] * t1;
}

// Y_j += c[j][k] * T   for j = 0..7
__global__ __launch_bounds__(256) void axpy8_k(const float* __restrict__ T,
                                               float* __restrict__ Y,
                                               const float* __restrict__ coef,
                                               int k) {
    size_t i = (size_t)blockIdx.x * blockDim.x + threadIdx.x;
    if (i >= (size_t)ND) return;
    float t = T[i];
    #pragma unroll
    for (int j = 0; j < JF; ++j)
        Y[(size_t)j * ND + i] += coef[j * KC + k] * t;
}

// U = |Y| over all 8 branch matrices
__global__ __launch_bounds__(256) void absmat_k(const float* __restrict__ Y,
                                                float* __restrict__ U) {
    size_t i = (size_t)blockIdx.x * blockDim.x + threadIdx.x;
    if (i >= (size_t)JF * ND) return;
    U[i] = fabsf(Y[i]);
}

// out[col] = mean over 2048 rows of (|X| or X) column `col`.  One block per column.
__global__ __launch_bounds__(256) void colmean_k(const float* __restrict__ X,
                                                 float* __restrict__ out,
                                                 int useAbs) {
    __shared__ float red[256];
    int col = blockIdx.x;
    float s = 0.0f;
    for (int r = threadIdx.x; r < NN; r += 256) {
        float v = X[(size_t)r * DF + col];
        s += useAbs ? fabsf(v) : v;
    }
    red[threadIdx.x] = s;
    __syncthreads();
    for (int st = 128; st > 0; st >>= 1) {
        if (threadIdx.x < st) red[threadIdx.x] += red[threadIdx.x + st];
        __syncthreads();
    }
    if (threadIdx.x == 0) out[col] = red[0] * (1.0f / (float)NN);
}

// ---------------------------------------------------------------------------
// Orchestration
// ---------------------------------------------------------------------------
extern "C" void kernel_launch(void* const* d_in, const int* in_sizes, int n_in,
                              void* d_out, int out_size, void* d_ws, size_t ws_size,
                              hipStream_t stream) {
    const float* W = (const float*)d_in[0];   // 2048 x 2048
    const float* f = (const float*)d_in[1];   // 2048 x 256
    float* out = (float*)d_out;               // 18688 floats

    // Workspace carve-up (all f32): ~56.6 MB
    float* Ls   = (float*)d_ws;               // shifted Laplacian   (NN*NN)
    float* deg  = Ls   + (size_t)NN * NN;     // NN
    float* dh   = deg  + NN;                  // NN
    float* coef = dh   + NN;                  // JF*KC
    float* b0   = coef + JF * KC;             // 3 Chebyshev T buffers, ND each
    float* b1   = b0 + ND;
    float* b2   = b1 + ND;
    float* Y    = b2 + ND;                    // 8 * ND accumulators
    float* U1   = Y  + (size_t)JF * ND;       // 8 * ND layer-1 outputs

    const int EPB = 256;
    dim3 ggrid(NN / 128, DF / 64);            // GEMM grid (16,4)

    // 1. Graph setup
    row_deg_k  <<<NN, 256, 0, stream>>>(W, deg, dh);
    build_lap_k<<<(int)(((size_t)NN * NN + EPB - 1) / EPB), EPB, 0, stream>>>(W, deg, dh, Ls);
    cheb_coeffs_k<<<JF, KC, 0, stream>>>(coef);

    // 2. Zeroth-order coefficients: mean of f over nodes
    colmean_k<<<DF, 256, 0, stream>>>(f, out, 0);

    float* bufs[3] = { b0, b1, b2 };

    // Chebyshev branch driver: Y_j = sum_k c[j][k] T_k(Ls) @ T0, T0 read-only.
    auto run_branch = [&](const float* T0) {
        // T1 = Ls @ T0
        gemm_f32_wmma<<<ggrid, 256, 0, stream>>>(Ls, T0, nullptr, b0, 1.0f, 0.0f);
        yinit_k<<<(ND + EPB - 1) / EPB, EPB, 0, stream>>>(T0, b0, Y, coef);
        const float* Tp = T0;
        const float* Tc = b0;
        int nx = 1;
        for (int k = 2; k < KC; ++k) {
            float* Tn = bufs[nx];
            // T_{k} = 2 * Ls @ T_{k-1} - T_{k-2}
            gemm_f32_wmma<<<ggrid, 256, 0, stream>>>(Ls, Tc, Tp, Tn, 2.0f, -1.0f);
            axpy8_k<<<(ND + EPB - 1) / EPB, EPB, 0, stream>>>(Tn, Y, coef, k);
            Tp = Tc; Tc = Tn; nx = (nx + 1) % 3;
        }
    };

    // 3. Layer 1: filter f with all 8 filters
    run_branch(f);
    for (int j = 0; j < JF; ++j)
        colmean_k<<<DF, 256, 0, stream>>>(Y + (size_t)j * ND, out + 256 + j * DF, 1);
    absmat_k<<<(int)(((size_t)JF * ND + EPB - 1) / EPB), EPB, 0, stream>>>(Y, U1);

    // 4. Layer 2: filter each |g_j f| with all 8 filters
    for (int b = 0; b < JF; ++b) {
        run_branch(U1 + (size_t)b * ND);
        for (int j = 0; j < JF; ++j)
            colmean_k<<<DF, 256, 0, stream>>>(Y + (size_t)j * ND,
                                              out + 256 + JF * DF + (b * JF + j) * DF, 1);
    }
}